// CLSAv4NoPosLoss_84774064488939
// MI455X (gfx1250) — compile-verified
//
#include <hip/hip_runtime.h>
#include <math.h>

// CDNA5 / gfx1250, wave32.
typedef __attribute__((ext_vector_type(2))) float v2f;
typedef __attribute__((ext_vector_type(8))) float v8f;

#define N_TOT   16384
#define K_DIM   128
#define HALF_N  8192
#define ROW_TILES (N_TOT / 16)           // 1024 tiles of 16 rows
#define NSTRIP    32                      // col strips per row tile
#define TILES_PER_STRIP (ROW_TILES / NSTRIP)          // 32 col tiles / strip
#define WAVES_PER_BLOCK 8
#define TILES_PER_WAVE  (TILES_PER_STRIP / WAVES_PER_BLOCK) // 4 col tiles / wave

// ---------------------------------------------------------------------------
// rowsum must start at zero (harness poisons d_ws).
__global__ __launch_bounds__(256) void k_zero(float* __restrict__ p) {
    p[blockIdx.x * 256 + threadIdx.x] = 0.0f;
}

// ---------------------------------------------------------------------------
// sq[i] = sum_k feats[i][k]^2 ; one wave per row (wave32: 4 floats/lane)
__global__ __launch_bounds__(256) void k_sq(const float* __restrict__ F,
                                            float* __restrict__ sq) {
    const int wave = threadIdx.x >> 5;
    const int lane = threadIdx.x & 31;
    const int row  = blockIdx.x * WAVES_PER_BLOCK + wave;
    const float4 f = *((const float4*)(F + (size_t)row * K_DIM) + lane);
    float s = f.x * f.x + f.y * f.y + f.z * f.z + f.w * f.w;
    s += __shfl_xor(s, 1);
    s += __shfl_xor(s, 2);
    s += __shfl_xor(s, 4);
    s += __shfl_xor(s, 8);
    s += __shfl_xor(s, 16);
    if (lane == 0) sq[row] = s;
}

// ---------------------------------------------------------------------------
// Fused Gram / Cauchy-q / row-sum / positive-pair kernel.
// grid = (ROW_TILES, NSTRIP); block = 256 threads = 8 waves.
// Each wave: fixed 16-row tile, 4 col tiles processed 2 at a time.
// Per tile pair: A fragments reused from registers, two B load streams,
// FOUR independent v_wmma_f32_16x16x4_f32 accumulation chains.
__global__ __launch_bounds__(256) void k_gram(const float* __restrict__ F,
                                              const float* __restrict__ sq,
                                              float* __restrict__ rowsum,
                                              float* __restrict__ qii) {
    const int rt    = blockIdx.x;            // row tile (0..1023)
    const int strip = blockIdx.y;            // col strip (0..31)
    const int wave  = threadIdx.x >> 5;
    const int lane  = threadIdx.x & 31;
    const int lmod  = lane & 15;             // 0..15
    const int lhi   = lane >> 4;             // 0 or 1
    const int kb    = 2 * lhi;               // per-lane K base (fp32 A layout)

    // ---- preload A fragments for this row tile (K = 0..127, 32 k-steps) ----
    // A 16x4 fp32 layout: lane L holds M = L%16, K = 2*(L/16) + {0,1}.
    const int arow = rt * 16 + lmod;
    const float* aptr = F + (size_t)arow * K_DIM + kb;
    v2f afrag[32];
#pragma unroll
    for (int t = 0; t < 32; ++t)
        afrag[t] = *(const v2f*)(aptr + 4 * t);

    // ||x||^2 for this lane's M rows (C/D layout: M = v + 8*(L/16))
    float sqr[8];
#pragma unroll
    for (int v = 0; v < 8; ++v)
        sqr[v] = sq[rt * 16 + v + 8 * lhi];

    v8f rs = {0.f, 0.f, 0.f, 0.f, 0.f, 0.f, 0.f, 0.f};  // partial row sums
    const int pc = (rt + HALF_N / 16) % ROW_TILES;       // positive-pair col tile

#pragma unroll 1
    for (int i = 0; i < TILES_PER_WAVE; i += 2) {
        const int ct0 = strip * TILES_PER_STRIP + wave * TILES_PER_WAVE + i;
        const int ct1 = ct0 + 1;
        const int bcol0 = ct0 * 16 + lmod;
        const int bcol1 = ct1 * 16 + lmod;
        // B 4x16 fp32 layout of F^T: lane L holds N = L%16, K = 2*(L/16)+{0,1}
        const float* bptr0 = F + (size_t)bcol0 * K_DIM + kb;
        const float* bptr1 = F + (size_t)bcol1 * K_DIM + kb;

        // Four independent accumulation chains (2 per col tile), A reused
        // from registers for both tiles; two B load streams for MLP.
        v8f acc0a = {0.f, 0.f, 0.f, 0.f, 0.f, 0.f, 0.f, 0.f};
        v8f acc0b = {0.f, 0.f, 0.f, 0.f, 0.f, 0.f, 0.f, 0.f};
        v8f acc1a = {0.f, 0.f, 0.f, 0.f, 0.f, 0.f, 0.f, 0.f};
        v8f acc1b = {0.f, 0.f, 0.f, 0.f, 0.f, 0.f, 0.f, 0.f};
#pragma unroll
        for (int t = 0; t < 16; ++t) {
            v2f b0e = *(const v2f*)(bptr0 + 8 * t);
            v2f b1e = *(const v2f*)(bptr1 + 8 * t);
            v2f b0o = *(const v2f*)(bptr0 + 8 * t + 4);
            v2f b1o = *(const v2f*)(bptr1 + 8 * t + 4);
            acc0a = __builtin_amdgcn_wmma_f32_16x16x4_f32(
                false, afrag[2 * t], false, b0e, (short)0, acc0a, false, false);
            acc1a = __builtin_amdgcn_wmma_f32_16x16x4_f32(
                false, afrag[2 * t], false, b1e, (short)0, acc1a, false, false);
            acc0b = __builtin_amdgcn_wmma_f32_16x16x4_f32(
                false, afrag[2 * t + 1], false, b0o, (short)0, acc0b, false, false);
            acc1b = __builtin_amdgcn_wmma_f32_16x16x4_f32(
                false, afrag[2 * t + 1], false, b1o, (short)0, acc1b, false, false);
        }

        const float sqc0 = sq[bcol0];        // ||y||^2 for this lane's N column
        const float sqc1 = sq[bcol1];
        float qv0[8], qv1[8];
#pragma unroll
        for (int v = 0; v < 8; ++v) {
            float g0  = acc0a[v] + acc0b[v];
            float g1  = acc1a[v] + acc1b[v];
            float d20 = fmaxf(sqr[v] + sqc0 - 2.0f * g0, 0.0f);
            float d21 = fmaxf(sqr[v] + sqc1 - 2.0f * g1, 0.0f);
            float q0 = 1.0f / (d20 + 1.0f);
            float q1 = 1.0f / (d21 + 1.0f);
            qv0[v] = q0;
            qv1[v] = q1;
            rs[v] += q0 + q1;
        }

        // positive pair q[i, (i+B)%n]: lands on tile diagonal of col tile pc.
        if (ct0 == pc) {
#pragma unroll
            for (int v = 0; v < 8; ++v)
                if (lmod == v + 8 * lhi)
                    qii[rt * 16 + v + 8 * lhi] = qv0[v];
        }
        if (ct1 == pc) {
#pragma unroll
            for (int v = 0; v < 8; ++v)
                if (lmod == v + 8 * lhi)
                    qii[rt * 16 + v + 8 * lhi] = qv1[v];
        }
    }

    // reduce row sums across the 16-lane halves (lanes 0-15: M=v, 16-31: M=v+8)
#pragma unroll
    for (int v = 0; v < 8; ++v) {
        float s = rs[v];
        s += __shfl_xor(s, 1);
        s += __shfl_xor(s, 2);
        s += __shfl_xor(s, 4);
        s += __shfl_xor(s, 8);
        rs[v] = s;
    }
    if (lane == 0 || lane == 16) {
#pragma unroll
        for (int v = 0; v < 8; ++v)
            atomicAdd(&rowsum[rt * 16 + v + 8 * lhi], rs[v]);
    }
}

// ---------------------------------------------------------------------------
// out = mean(-log(qii)) + mean(log(rowsum)) * S_HAT   (S_HAT == 1.0)
__global__ __launch_bounds__(256) void k_final(const float* __restrict__ rowsum,
                                               const float* __restrict__ qii,
                                               float* __restrict__ out) {
    __shared__ float red[256];
    float acc = 0.0f;
    for (int i = threadIdx.x; i < N_TOT; i += 256)
        acc += -logf(qii[i]) + logf(rowsum[i]) * 1.0f;
    red[threadIdx.x] = acc;
    __syncthreads();
    for (int s = 128; s > 0; s >>= 1) {
        if (threadIdx.x < s) red[threadIdx.x] += red[threadIdx.x + s];
        __syncthreads();
    }
    if (threadIdx.x == 0) out[0] = red[0] / (float)N_TOT;
}

// ---------------------------------------------------------------------------
extern "C" void kernel_launch(void* const* d_in, const int* in_sizes, int n_in,
                              void* d_out, int out_size, void* d_ws, size_t ws_size,
                              hipStream_t stream) {
    const float* feats = (const float*)d_in[0];
    // d_in[1] (idx, int64) is unused by the reference.

    float* sq     = (float*)d_ws;        // 16384 floats
    float* rowsum = sq + N_TOT;          // 16384 floats
    float* qii    = rowsum + N_TOT;      // 16384 floats

    k_zero<<<N_TOT / 256, 256, 0, stream>>>(rowsum);
    k_sq<<<N_TOT / WAVES_PER_BLOCK, 256, 0, stream>>>(feats, sq);

    dim3 grid(ROW_TILES, NSTRIP);
    k_gram<<<grid, 256, 0, stream>>>(feats, sq, rowsum, qii);

    k_final<<<1, 256, 0, stream>>>(rowsum, qii, (float*)d_out);
}